// DeformConv2d_12326556140206
// MI455X (gfx1250) — compile-verified
//
#include <hip/hip_runtime.h>
#include <cstddef>

typedef __attribute__((ext_vector_type(2))) float v2f;
typedef __attribute__((ext_vector_type(4))) float v4f;
typedef __attribute__((ext_vector_type(8))) float v8f;

#define BATCH 8
#define CIN   64
#define OCN   128
#define HH    128
#define WW    128
#define HP    130
#define WP    130
#define KN    9
#define KTOT  576          // CIN * KN
#define ASTRIDE 578        // padded LDS row stride (even, !=0 mod 64)
#define TILE_M 64          // pixels per block (four 16-row WMMA tiles)

#define XP_FLOATS ((size_t)BATCH * HP * WP * CIN)          // 8,652,800
#define XP_BYTES  (XP_FLOATS * 4)                          // 34,611,200

// ---------------------------------------------------------------------------
// Kernel 1: zero-fill the padded NHWC image buffer
// ---------------------------------------------------------------------------
__global__ void fill_zero_kernel(v4f* __restrict__ p, int n4) {
    int i = blockIdx.x * blockDim.x + threadIdx.x;
    if (i < n4) p[i] = (v4f){0.f, 0.f, 0.f, 0.f};
}

// ---------------------------------------------------------------------------
// Kernel 2: NCHW -> zero-padded NHWC via LDS transpose.
// One block per (b, h) row: loads 64c x 128w coalesced, stores 128w x 64c
// coalesced into the interior of xp.
// ---------------------------------------------------------------------------
__global__ __launch_bounds__(256) void nchw_to_padded_nhwc(
    const float* __restrict__ x, float* __restrict__ xp)
{
    __shared__ float t[CIN * 129];          // 129 stride -> conflict-free
    int bh = blockIdx.x;                    // 0 .. 8*128-1
    int b = bh >> 7, h = bh & 127;
    const float* src = x + ((size_t)b << 20) + (h << 7);   // b*64*16384 + h*128
    #pragma unroll
    for (int i = 0; i < 32; ++i) {
        int idx = threadIdx.x + (i << 8);
        int c = idx >> 7, w = idx & 127;
        t[c * 129 + w] = src[((size_t)c << 14) + w];
    }
    __syncthreads();
    float* dst = xp + (((size_t)b * HP + (h + 1)) * WP + 1) * CIN;
    #pragma unroll
    for (int i = 0; i < 32; ++i) {
        int idx = threadIdx.x + (i << 8);
        int w = idx >> 6, c = idx & 63;
        dst[(size_t)w * CIN + c] = t[c * 129 + w];
    }
}

// ---------------------------------------------------------------------------
// Kernel 3: offset-predicting 3x3 conv (18 output channels), direct form.
// One thread per pixel; weights staged in LDS (broadcast reads).
// offs layout: [b][18][128][128] (NCHW, matches reference indexing)
// ---------------------------------------------------------------------------
__global__ __launch_bounds__(256) void offset_conv_kernel(
    const float* __restrict__ xp, const float* __restrict__ wp,
    const float* __restrict__ bp, float* __restrict__ offs)
{
    __shared__ float wl[18 * KTOT];
    for (int i = threadIdx.x; i < 18 * KTOT; i += 256) wl[i] = wp[i];
    __syncthreads();

    int pix = blockIdx.x * 256 + threadIdx.x;   // 0 .. 131071
    int b = pix >> 14;
    int hw = pix & 16383;
    int h = hw >> 7, w = hw & 127;

    float acc[18];
    #pragma unroll
    for (int oc = 0; oc < 18; ++oc) acc[oc] = bp[oc];

    const float* base = xp + (size_t)b * (HP * WP * CIN);
    for (int kh = 0; kh < 3; ++kh) {
        for (int kw = 0; kw < 3; ++kw) {
            int tap = kh * 3 + kw;
            const float* cell = base + ((size_t)((h + kh) * WP + (w + kw))) * CIN;
            for (int c = 0; c < CIN; c += 4) {
                v4f v = *(const v4f*)(cell + c);
                #pragma unroll
                for (int j = 0; j < 4; ++j) {
                    float xv = v[j];
                    #pragma unroll
                    for (int oc = 0; oc < 18; ++oc)
                        acc[oc] = fmaf(xv, wl[oc * KTOT + (c + j) * KN + tap], acc[oc]);
                }
            }
        }
    }
    float* ob = offs + ((size_t)b * 18 << 14) + hw;
    #pragma unroll
    for (int oc = 0; oc < 18; ++oc) ob[(size_t)oc << 14] = acc[oc];
}

// ---------------------------------------------------------------------------
// Kernel 4: deformable gather -> im2col A-tile in LDS -> fp32 WMMA GEMM.
// Block = 256 threads (8 waves), one 64-pixel M-tile (64 consecutive w).
// Phase 1: 576 (pixel,tap) pairs over 256 threads: bilinear corners/weights
//          once per pair, float4 channel gathers, fill A[64][576] in LDS.
// Phase 2: wave v computes out-channels [16v,16v+16) for FOUR 16-row halves,
//          sharing each B fragment across four v_wmma_f32_16x16x4_f32 chains.
//          Outer 64-K chunk loop kept rolled (compact hot loop, I-cache
//          friendly); next B chunk prefetched via global_prefetch_b8.
// Phase 3: C-fragment rows map to consecutive w -> aligned b128 stores.
// LDS: 64*578*4 = 147,968 B -> exactly 2 blocks per 320 KB WGP, so one
// block's gather (VMEM/VALU) overlaps the other's WMMA phase.
// ---------------------------------------------------------------------------
__global__ __launch_bounds__(256) void deform_gemm_kernel(
    const float* __restrict__ xp,     // [8][130][130][64]
    const float* __restrict__ offs,   // [8][18][128][128]
    const float* __restrict__ wconv,  // [128][576] row-major (o, c*9+tap)
    float* __restrict__ out)          // [8][128][128][128]
{
    __shared__ float A[TILE_M * ASTRIDE];  // 147,968 B

    int tid = threadIdx.x;
    int pix_base = blockIdx.x << 6;        // 64 pixels per tile
    int b = pix_base >> 14;
    int hw = pix_base & 16383;
    int h = hw >> 7;
    int w0 = hw & 127;                     // 0 or 64

    // ---- Phase 1: build A tile -------------------------------------------
    for (int pair = tid; pair < TILE_M * KN; pair += 256) {
        int m = pair / 9;                  // pixel within tile (0..63)
        int n = pair % 9;                  // tap
        int wc = w0 + m;
        float dx = (float)(n / 3 - 1);
        float dy = (float)(n % 3 - 1);
        const float* offb = offs + ((size_t)b * 18 << 14) + (h << 7) + wc;
        float ox = offb[(size_t)n << 14];
        float oy = offb[(size_t)(n + 9) << 14];
        float px = (float)h  + dx + ox;
        float py = (float)wc + dy + oy;
        float fx = floorf(px), fy = floorf(py);
        const float hi = (float)(HP - 1);
        float xlt = fminf(fmaxf(fx,       0.f), hi);
        float ylt = fminf(fmaxf(fy,       0.f), hi);
        float xrb = fminf(fmaxf(fx + 1.f, 0.f), hi);
        float yrb = fminf(fmaxf(fy + 1.f, 0.f), hi);
        float pxc = fminf(fmaxf(px, 0.f), hi);
        float pyc = fminf(fmaxf(py, 0.f), hi);
        float gx_l = 1.f + xlt - pxc;
        float gx_r = 1.f - xrb + pxc;
        float gy_l = 1.f + ylt - pyc;
        float gy_r = 1.f - yrb + pyc;
        float g_lt = gx_l * gy_l;
        float g_rb = gx_r * gy_r;
        float g_lb = gx_l * gy_r;
        float g_rt = gx_r * gy_l;
        int ixl = (int)xlt, iyl = (int)ylt, ixr = (int)xrb, iyr = (int)yrb;
        const float* base = xp + (size_t)b * (HP * WP * CIN);
        const float* p_lt = base + (size_t)(ixl * WP + iyl) * CIN;
        const float* p_rb = base + (size_t)(ixr * WP + iyr) * CIN;
        const float* p_lb = base + (size_t)(ixl * WP + iyr) * CIN;
        const float* p_rt = base + (size_t)(ixr * WP + iyl) * CIN;
        float* arow = A + m * ASTRIDE + n;
        for (int c = 0; c < CIN; c += 4) {
            v4f vlt = *(const v4f*)(p_lt + c);
            v4f vrb = *(const v4f*)(p_rb + c);
            v4f vlb = *(const v4f*)(p_lb + c);
            v4f vrt = *(const v4f*)(p_rt + c);
            v4f v = g_lt * vlt + g_rb * vrb + g_lb * vlb + g_rt * vrt;
            arow[(c + 0) * KN] = v.x;
            arow[(c + 1) * KN] = v.y;
            arow[(c + 2) * KN] = v.z;
            arow[(c + 3) * KN] = v.w;
        }
    }
    __syncthreads();

    // ---- Phase 2: WMMA GEMM, four M-quarters share each B fragment -------
    int wv   = tid >> 5;                   // wave -> 16-channel slice
    int lane = tid & 31;
    int mrow = lane & 15;
    int hi2  = lane >> 4;                  // K split: lanes 16-31 hold K+2
    const float* Aptr0 = A + mrow * ASTRIDE + 2 * hi2;
    const float* Aptr1 = Aptr0 + 16 * ASTRIDE;
    const float* Aptr2 = Aptr0 + 32 * ASTRIDE;
    const float* Aptr3 = Aptr0 + 48 * ASTRIDE;
    const float* Bptr  = wconv + (size_t)(wv * 16 + mrow) * KTOT + 2 * hi2;
    v8f acc0 = {};
    v8f acc1 = {};
    v8f acc2 = {};
    v8f acc3 = {};
    #pragma unroll 1
    for (int kko = 0; kko < KTOT; kko += 64) {
        __builtin_prefetch(Bptr + kko + 64, 0, 1);   // next B chunk (speculative)
        #pragma unroll
        for (int kk = kko; kk < kko + 64; kk += 4) {
            v2f bb = *(const v2f*)(Bptr  + kk);
            v2f a0 = *(const v2f*)(Aptr0 + kk);
            v2f a1 = *(const v2f*)(Aptr1 + kk);
            v2f a2 = *(const v2f*)(Aptr2 + kk);
            v2f a3 = *(const v2f*)(Aptr3 + kk);
            acc0 = __builtin_amdgcn_wmma_f32_16x16x4_f32(
                false, a0, false, bb, (short)0, acc0, false, false);
            acc1 = __builtin_amdgcn_wmma_f32_16x16x4_f32(
                false, a1, false, bb, (short)0, acc1, false, false);
            acc2 = __builtin_amdgcn_wmma_f32_16x16x4_f32(
                false, a2, false, bb, (short)0, acc2, false, false);
            acc3 = __builtin_amdgcn_wmma_f32_16x16x4_f32(
                false, a3, false, bb, (short)0, acc3, false, false);
        }
    }

    // ---- Phase 3: store (consecutive regs = consecutive w pixels) --------
    int o  = wv * 16 + (lane & 15);
    int mb = hi2 * 8;
    float* op = out + ((size_t)(b * OCN + o) << 14) + (h << 7) + w0 + mb;
    *(v4f*)(op)      = (v4f){acc0[0], acc0[1], acc0[2], acc0[3]};
    *(v4f*)(op + 4)  = (v4f){acc0[4], acc0[5], acc0[6], acc0[7]};
    *(v4f*)(op + 16) = (v4f){acc1[0], acc1[1], acc1[2], acc1[3]};
    *(v4f*)(op + 20) = (v4f){acc1[4], acc1[5], acc1[6], acc1[7]};
    *(v4f*)(op + 32) = (v4f){acc2[0], acc2[1], acc2[2], acc2[3]};
    *(v4f*)(op + 36) = (v4f){acc2[4], acc2[5], acc2[6], acc2[7]};
    *(v4f*)(op + 48) = (v4f){acc3[0], acc3[1], acc3[2], acc3[3]};
    *(v4f*)(op + 52) = (v4f){acc3[4], acc3[5], acc3[6], acc3[7]};
}

// ---------------------------------------------------------------------------
extern "C" void kernel_launch(void* const* d_in, const int* in_sizes, int n_in,
                              void* d_out, int out_size, void* d_ws, size_t ws_size,
                              hipStream_t stream) {
    const float* x      = (const float*)d_in[0];   // (8,64,128,128)
    const float* w_conv = (const float*)d_in[1];   // (128,64,3,3)
    const float* w_p    = (const float*)d_in[2];   // (18,64,3,3)
    const float* b_p    = (const float*)d_in[3];   // (18,)
    float* out = (float*)d_out;                    // (8,128,128,128)

    float* xp   = (float*)d_ws;                          // padded NHWC image
    float* offs = (float*)((char*)d_ws + XP_BYTES);      // predicted offsets

    int n4 = (int)(XP_FLOATS / 4);
    fill_zero_kernel<<<(n4 + 255) / 256, 256, 0, stream>>>((v4f*)xp, n4);
    nchw_to_padded_nhwc<<<BATCH * HH, 256, 0, stream>>>(x, xp);
    offset_conv_kernel<<<(BATCH * HH * WW) / 256, 256, 0, stream>>>(xp, w_p, b_p, offs);
    deform_gemm_kernel<<<(BATCH * HH * WW) / TILE_M, 256, 0, stream>>>(xp, offs, w_conv, out);
}